// HEALPix_Sampler_85023172592644
// MI455X (gfx1250) — compile-verified
//
#include <hip/hip_runtime.h>

typedef float v2f __attribute__((ext_vector_type(2)));
typedef float v8f __attribute__((ext_vector_type(8)));

#define B_    2
#define C_    2
#define T_    4
#define BCT   16          // B*C*T
#define NSRC  49152
#define NTGT  12288
#define D_    128
#define WSTR  132         // LDS row stride for W (pad 4 floats -> conflict-free b64 reads)

// ---------------- zero workspace ----------------
__global__ void zero_f4_kernel(float4* __restrict__ p, size_t n4) {
  size_t i = (size_t)blockIdx.x * blockDim.x + threadIdx.x;
  size_t stride = (size_t)gridDim.x * blockDim.x;
  float4 z; z.x = 0.f; z.y = 0.f; z.z = 0.f; z.w = 0.f;
  for (; i < n4; i += stride) p[i] = z;
}

// ---------------- segment counts ----------------
__global__ void count_kernel(const int* __restrict__ pm, float* __restrict__ cnt) {
  int i = blockIdx.x * blockDim.x + threadIdx.x;
  if (i < NSRC) atomicAdd(&cnt[pm[i]], 1.0f);
}

// ---------------- scatter-add pooling ----------------
// one wave per (bct, n) row of 128 floats; lane handles 4 contiguous floats.
// sums (96MB) is L2-resident on MI455X (192MB L2): atomics never hit HBM.
__global__ void scatter_kernel(const float* __restrict__ x,
                               const int* __restrict__ pm,
                               float* __restrict__ sums) {
  size_t tid  = (size_t)blockIdx.x * blockDim.x + threadIdx.x;
  size_t wave = tid >> 5;                // == bct*NSRC + n (n fastest)
  int lane = (int)(threadIdx.x & 31);
  int n    = (int)(wave % NSRC);
  int bct  = (int)(wave / NSRC);
  const float4 v = ((const float4*)(x + wave * (size_t)D_))[lane];
  int p = pm[n];
  float* dst = sums + ((size_t)p * BCT + bct) * D_ + lane * 4;
  atomicAdd(dst + 0, v.x);
  atomicAdd(dst + 1, v.y);
  atomicAdd(dst + 2, v.z);
  atomicAdd(dst + 3, v.w);
}

// ---------------- mean + 1x1 conv via WMMA f32 16x16x4 ----------------
// Each wave: 16(segments) x 128(out channels) tile for fixed bct.
// W staged in LDS via the gfx1250 async-copy path; B-frags via conflict-free ds_load_b64.
__global__ void __launch_bounds__(256) gemm_kernel(
    const float* __restrict__ sums,
    const float* __restrict__ cnt,
    const float* __restrict__ W,
    const float* __restrict__ bias,
    float* __restrict__ out) {
  __shared__ float wlds[D_ * WSTR];      // 67,584 B

  // ---- stage W[128][128] -> LDS (row stride WSTR) with async copies ----
  // 4096 b128 chunks total; 256 threads x 16 chunks each.
  // NOTE: the LDS operand is derived from &wlds[...] so that (a) its low 32
  // bits are the wave-relative LDS byte offset (LDS aperture = addr[31:0]),
  // and (b) wlds escapes into the asm, making the "memory" clobber alias it.
  {
    unsigned t = threadIdx.x;
#pragma unroll
    for (int it = 0; it < 16; ++it) {
      unsigned ch = (unsigned)it * 256u + t;
      unsigned r  = ch >> 5;             // W row
      unsigned c4 = (ch & 31u) * 4u;     // float col (16B chunk)
      unsigned lds_off = (unsigned)(uintptr_t)(const void*)&wlds[r * WSTR + c4];
      unsigned long long ga = (unsigned long long)(const void*)(W + (size_t)r * D_ + c4);
      asm volatile("global_load_async_to_lds_b128 %0, %1, off"
                   :: "v"(lds_off), "v"(ga) : "memory");
    }
    asm volatile("s_wait_asynccnt 0x0" ::: "memory");
  }
  __syncthreads();

  unsigned tid  = blockIdx.x * blockDim.x + threadIdx.x;
  unsigned wave = tid >> 5;
  unsigned lane = threadIdx.x & 31;
  const unsigned SEGT = NTGT / 16;       // 768 tiles per bct
  unsigned bct  = wave / SEGT;
  unsigned seg0 = (wave % SEGT) * 16;
  unsigned m    = lane & 15;             // A-row / B,D-column for this lane
  unsigned koff = (lane >> 4) * 2;       // lanes 0-15: K,K+1; lanes 16-31: K+2,K+3

  // fused mean: per-lane reciprocal count for A-row (seg0+m)
  float c  = cnt[seg0 + m];
  float rc = 1.0f / fmaxf(c, 1e-6f);

  // ---- preload all 32 A-fragments (b64 each), clause-batched, one wait ----
  const float* arow = sums + ((size_t)(seg0 + m) * BCT + bct) * D_ + koff;
  v2f afrag[32];
#pragma unroll
  for (int ks = 0; ks < 32; ++ks)
    afrag[ks] = *(const v2f*)(arow + 4 * ks);

  // B-frag base in LDS: wlds[n*WSTR + k + koff], n = nt*16 + m
  const float* bb = &wlds[(size_t)m * WSTR + koff];

  v8f acc[8] = {};
#pragma unroll
  for (int ks = 0; ks < 32; ++ks) {
    v2f a = afrag[ks] * rc;
    int k = ks * 4;
#pragma unroll
    for (int nt = 0; nt < 8; ++nt) {
      v2f bf = *(const v2f*)(bb + nt * (16 * WSTR) + k);   // ds_load_b64, conflict-free
      acc[nt] = __builtin_amdgcn_wmma_f32_16x16x4_f32(
          /*neg_a=*/false, a, /*neg_b=*/false, bf,
          /*c_mod=*/(short)0, acc[nt], /*reuse_a=*/false, /*reuse_b=*/false);
    }
  }

  // D layout: VGPR r -> row M=r (lanes 0-15) / M=r+8 (lanes 16-31), N=lane&15
  unsigned mrow = (lane >> 4) * 8;
#pragma unroll
  for (int nt = 0; nt < 8; ++nt) {
    unsigned e  = nt * 16 + m;
    float bv = bias[e];
#pragma unroll
    for (int r = 0; r < 8; ++r) {
      unsigned seg = seg0 + r + mrow;
      out[((size_t)bct * NTGT + seg) * D_ + e] = acc[nt][r] + bv;
    }
  }
}

extern "C" void kernel_launch(void* const* d_in, const int* in_sizes, int n_in,
                              void* d_out, int out_size, void* d_ws, size_t ws_size,
                              hipStream_t stream) {
  const float* x    = (const float*)d_in[0];   // [B,C,T,NSRC,D] fp32
  const int*   pm   = (const int*)d_in[1];     // [NSRC] int32
  const float* W    = (const float*)d_in[2];   // [D,D] fp32
  const float* bias = (const float*)d_in[3];   // [D] fp32
  float* out = (float*)d_out;

  float* sums = (float*)d_ws;                            // [NTGT, BCT, D]
  float* cnt  = sums + (size_t)NTGT * BCT * D_;          // [NTGT]

  // 1) zero sums + counts (harness poisons d_ws)
  size_t zfloats = (size_t)NTGT * BCT * D_ + NTGT;       // multiples of 4
  zero_f4_kernel<<<4096, 256, 0, stream>>>((float4*)d_ws, zfloats / 4);

  // 2) segment counts
  count_kernel<<<(NSRC + 255) / 256, 256, 0, stream>>>(pm, cnt);

  // 3) scatter-add pooling: BCT*NSRC waves, 8 waves/block
  {
    size_t waves = (size_t)BCT * NSRC;                   // 786432
    unsigned blocks = (unsigned)(waves / 8);             // 98304, exact
    scatter_kernel<<<blocks, 256, 0, stream>>>(x, pm, sums);
  }

  // 4) mean + GEMM + bias: BCT*(NTGT/16) = 12288 waves, exact grid (EXEC all-ones)
  {
    unsigned waves = BCT * (NTGT / 16);                  // 12288
    unsigned blocks = waves / 8;                         // 1536, exact
    gemm_kernel<<<blocks, 256, 0, stream>>>(sums, cnt, W, bias, out);
  }
}